// DialogueGraphModel_4355096838650
// MI455X (gfx1250) — compile-verified
//
#include <hip/hip_runtime.h>
#include <math.h>

#define HH 8
#define DHH 16
#define NEG_SLOPE 0.2f

typedef float v2f __attribute__((ext_vector_type(2)));
typedef float v8f __attribute__((ext_vector_type(8)));

// ---- order-preserving float<->uint encoding for atomicMax-based segment max ----
__device__ __forceinline__ unsigned enc_f32(float x) {
    unsigned u = __float_as_uint(x);
    return (u >> 31) ? ~u : (u | 0x80000000u);
}
__device__ __forceinline__ float dec_f32(unsigned e) {
    return (e >> 31) ? __uint_as_float(e & 0x7FFFFFFFu) : __uint_as_float(~e);
}

// ---- async copy of one 16x128 fp32 A tile into LDS (row stride 132 floats) ----
// 512 chunks of 16B; 256 threads issue exactly 2 GLOBAL_LOAD_ASYNC_TO_LDS_B128 each
__device__ __forceinline__ void async_copy_tile(const float* __restrict__ src, int row0,
                                                float* sbuf, int D, int tid) {
    for (int c = tid; c < 512; c += 256) {
        int row = c >> 5;
        int chunk = c & 31;
        unsigned lds = (unsigned)(size_t)(sbuf + row * 132 + chunk * 4);
        unsigned long long g =
            (unsigned long long)(size_t)(src + (size_t)(row0 + row) * D + chunk * 4);
        asm volatile("global_load_async_to_lds_b128 %0, %1, off"
                     :: "v"(lds), "v"(g) : "memory");
    }
}

// ---------------- Kernel 1: masked mean pool + speaker embedding ----------------
__global__ void pool_kernel(const float* __restrict__ feat, const int* __restrict__ mask,
                            const int* __restrict__ spk_id, const float* __restrict__ spk_emb,
                            float* __restrict__ x, int N, int L, int D) {
    int n = blockIdx.x;
    int d = threadIdx.x;
    const float* fp = feat + (size_t)n * L * D + d;
    float s = 0.f;
    int cnt = 0;
    for (int l = 0; l < L; ++l) {
        if (mask[n * L + l]) { s += fp[(size_t)l * D]; ++cnt; }
    }
    float denom = cnt > 0 ? (float)cnt : 1.f;
    x[(size_t)n * D + d] = s / denom + spk_emb[(size_t)spk_id[n] * D + d];
}

// ---------------- Kernel 2: WMMA GEMMs  Y = X @ W  (W_r[0..R-1] and W_q) ----------------
// 256 threads = 8 waves; wave w owns columns [16w,16w+16); B fully register-resident;
// A tiles double-buffered in LDS via async loads.
__global__ void gemm_wmma_kernel(const float* __restrict__ X,
                                 const float* __restrict__ Wr, const float* __restrict__ Wq,
                                 float* __restrict__ Hr, float* __restrict__ Q,
                                 int N, int D, int R) {
    __shared__ float sA[2][16 * 132];
    int wsel = blockIdx.y;
    const float* W = (wsel < R) ? (Wr + (size_t)wsel * D * D) : Wq;
    float* Y = (wsel < R) ? (Hr + (size_t)wsel * N * D) : Q;

    int lane = threadIdx.x & 31;
    int wv   = threadIdx.x >> 5;
    int half = lane >> 4;        // 0: K pair {0,1}, 1: K pair {2,3}
    int l15  = lane & 15;
    int col  = (wv << 4) + l15;  // this lane's B/C column

    // preload this wave's B panel (16 cols x K=128) into registers
    v2f breg[32];
#pragma unroll
    for (int kk = 0; kk < 32; ++kk) {
        int k = (kk << 2) + (half << 1);
        breg[kk].x = W[(size_t)k * D + col];
        breg[kk].y = W[(size_t)(k + 1) * D + col];
    }

    int nTiles = N >> 4;
    int stride = gridDim.x;
    int t0 = blockIdx.x;
    if (t0 >= nTiles) return;

    async_copy_tile(X, t0 << 4, sA[0], D, threadIdx.x);
    int pb = 0;
    for (int t = t0; t < nTiles; t += stride) {
        int tn = t + stride;
        async_copy_tile(X, (tn < nTiles ? tn : t) << 4, sA[pb ^ 1], D, threadIdx.x);
        asm volatile("s_wait_asynccnt 0x2" ::: "memory");  // tile t resident (2 newest pending)
        __syncthreads();

        const float* Ab = sA[pb] + l15 * 132 + (half << 1);
        v8f acc = {};
#pragma unroll
        for (int kk = 0; kk < 32; ++kk) {
            v2f a; a.x = Ab[kk * 4]; a.y = Ab[kk * 4 + 1];
            acc = __builtin_amdgcn_wmma_f32_16x16x4_f32(false, a, false, breg[kk],
                                                        (short)0, acc, false, false);
        }
        int row0 = t << 4;
        float* Yp = Y + (size_t)(row0 + (half << 3)) * D + col;
#pragma unroll
        for (int v = 0; v < 8; ++v) Yp[(size_t)v * D] = acc[v];
        __syncthreads();
        pb ^= 1;
    }
}

// ---------------- Kernel 3: per-node score components ----------------
__global__ void node_score_kernel(const float* __restrict__ Hr, const float* __restrict__ Q,
                                  const float* __restrict__ a_src, const float* __restrict__ a_dst,
                                  float* __restrict__ Ssrc, float* __restrict__ Sdst,
                                  int N, int D, int R) {
    int t = blockIdx.x * blockDim.x + threadIdx.x;
    if (t >= N * HH) return;
    int n = t / HH, h = t % HH;

    const float4* ad = (const float4*)(a_dst + h * DHH);
    const float4* qp = (const float4*)(Q + (size_t)n * D + h * DHH);
    float s = 0.f;
#pragma unroll
    for (int i = 0; i < 4; ++i) {
        float4 a = ad[i], b = qp[i];
        s += a.x * b.x + a.y * b.y + a.z * b.z + a.w * b.w;
    }
    Sdst[t] = s;

    const float4* as = (const float4*)(a_src + h * DHH);
    for (int r = 0; r < R; ++r) {
        const float4* hp = (const float4*)(Hr + ((size_t)r * N + n) * D + h * DHH);
        float ss = 0.f;
#pragma unroll
        for (int i = 0; i < 4; ++i) {
            float4 a = as[i], b = hp[i];
            ss += a.x * b.x + a.y * b.y + a.z * b.z + a.w * b.w;
        }
        Ssrc[((size_t)r * N + n) * HH + h] = ss;
    }
}

__device__ __forceinline__ float edge_score(const float* Ssrc, const float* Sdst,
                                            int r, int src, int tgt, int h, int N) {
    float s = Ssrc[((size_t)r * N + src) * HH + h] + Sdst[(size_t)tgt * HH + h];
    return s > 0.f ? s : NEG_SLOPE * s;
}

// ---------------- Kernel 4: segment max over incoming edges ----------------
__global__ void edge_max_kernel(const int* __restrict__ ei, const int* __restrict__ et,
                                const float* __restrict__ Ssrc, const float* __restrict__ Sdst,
                                unsigned* __restrict__ Menc, int E, int N) {
    int idx = blockIdx.x * blockDim.x + threadIdx.x;
    if (idx >= E * HH) return;
    int e = idx / HH, h = idx % HH;
    int src = ei[e], tgt = ei[E + e], r = et[e];
    float s = edge_score(Ssrc, Sdst, r, src, tgt, h, N);
    atomicMax(Menc + (size_t)tgt * HH + h, enc_f32(s));
}

// ---------------- Kernel 5: softmax denominator ----------------
__global__ void edge_den_kernel(const int* __restrict__ ei, const int* __restrict__ et,
                                const float* __restrict__ Ssrc, const float* __restrict__ Sdst,
                                const unsigned* __restrict__ Menc, float* __restrict__ Den,
                                int E, int N) {
    int idx = blockIdx.x * blockDim.x + threadIdx.x;
    if (idx >= E * HH) return;
    int e = idx / HH, h = idx % HH;
    int src = ei[e], tgt = ei[E + e], r = et[e];
    float s = edge_score(Ssrc, Sdst, r, src, tgt, h, N);
    float m = dec_f32(Menc[(size_t)tgt * HH + h]);
    atomicAdd(Den + (size_t)tgt * HH + h, expf(s - m));
}

// ---------------- Kernel 6: weighted message aggregation (gather + atomic scatter) ----------------
__global__ void edge_agg_kernel(const int* __restrict__ ei, const int* __restrict__ et,
                                const float* __restrict__ Ssrc, const float* __restrict__ Sdst,
                                const unsigned* __restrict__ Menc, const float* __restrict__ Den,
                                const float* __restrict__ Hr, float* __restrict__ Agg,
                                int E, int N, int D) {
    int tid = blockIdx.x * blockDim.x + threadIdx.x;
    if (tid >= E * 32) return;
    int e = tid >> 5;
    int ln = tid & 31;
    int d = ln << 2;
    int h = d >> 4;
    int src = ei[e], tgt = ei[E + e], r = et[e];
    size_t sh = (size_t)tgt * HH + h;
    float s = edge_score(Ssrc, Sdst, r, src, tgt, h, N);
    float alpha = expf(s - dec_f32(Menc[sh])) / (Den[sh] + 1e-9f);
    const float4 msg = *(const float4*)(Hr + ((size_t)r * N + src) * D + d);
    float* ap = Agg + (size_t)tgt * D + d;
    atomicAdd(ap + 0, msg.x * alpha);
    atomicAdd(ap + 1, msg.y * alpha);
    atomicAdd(ap + 2, msg.z * alpha);
    atomicAdd(ap + 3, msg.w * alpha);
}

// ---------------- Kernel 7: out = elu(x + agg @ W_out + b_out) ----------------
__global__ void gemm_out_kernel(const float* __restrict__ A, const float* __restrict__ Wg,
                                const float* __restrict__ Xres, const float* __restrict__ bias,
                                float* __restrict__ Out, int N, int D) {
    __shared__ float sA[2][16 * 132];
    int lane = threadIdx.x & 31;
    int wv   = threadIdx.x >> 5;
    int half = lane >> 4;
    int l15  = lane & 15;
    int col  = (wv << 4) + l15;

    v2f breg[32];
#pragma unroll
    for (int kk = 0; kk < 32; ++kk) {
        int k = (kk << 2) + (half << 1);
        breg[kk].x = Wg[(size_t)k * D + col];
        breg[kk].y = Wg[(size_t)(k + 1) * D + col];
    }
    float bcol = bias[col];

    int nTiles = N >> 4;
    int stride = gridDim.x;
    int t0 = blockIdx.x;
    if (t0 >= nTiles) return;

    async_copy_tile(A, t0 << 4, sA[0], D, threadIdx.x);
    int pb = 0;
    for (int t = t0; t < nTiles; t += stride) {
        int tn = t + stride;
        async_copy_tile(A, (tn < nTiles ? tn : t) << 4, sA[pb ^ 1], D, threadIdx.x);
        asm volatile("s_wait_asynccnt 0x2" ::: "memory");
        __syncthreads();

        const float* Ab = sA[pb] + l15 * 132 + (half << 1);
        v8f acc = {};
#pragma unroll
        for (int kk = 0; kk < 32; ++kk) {
            v2f a; a.x = Ab[kk * 4]; a.y = Ab[kk * 4 + 1];
            acc = __builtin_amdgcn_wmma_f32_16x16x4_f32(false, a, false, breg[kk],
                                                        (short)0, acc, false, false);
        }
        int row0 = t << 4;
#pragma unroll
        for (int v = 0; v < 8; ++v) {
            int row = row0 + (half << 3) + v;
            float z = acc[v] + Xres[(size_t)row * D + col] + bcol;
            Out[(size_t)row * D + col] = z > 0.f ? z : expm1f(z);
        }
        __syncthreads();
        pb ^= 1;
    }
}

extern "C" void kernel_launch(void* const* d_in, const int* in_sizes, int n_in,
                              void* d_out, int out_size, void* d_ws, size_t ws_size,
                              hipStream_t stream) {
    const float* feat    = (const float*)d_in[0];
    const int*   mask    = (const int*)d_in[1];
    const int*   ei      = (const int*)d_in[2];   // [2,E]
    const int*   et      = (const int*)d_in[3];
    const int*   spk     = (const int*)d_in[4];
    const float* spk_emb = (const float*)d_in[5];
    const float* W_r     = (const float*)d_in[6];
    const float* W_q     = (const float*)d_in[7];
    const float* a_src   = (const float*)d_in[8];
    const float* a_dst   = (const float*)d_in[9];
    const float* W_out   = (const float*)d_in[10];
    const float* b_out   = (const float*)d_in[11];

    int N = in_sizes[4];
    int D = in_sizes[11];
    int L = in_sizes[1] / N;
    int E = in_sizes[3];
    int R = in_sizes[6] / (D * D);
    float* out = (float*)d_out;

    // workspace carve-up (~190 MB)
    float* p   = (float*)d_ws;
    float* x   = p; p += (size_t)N * D;
    float* q   = p; p += (size_t)N * D;
    float* agg = p; p += (size_t)N * D;
    float* hr  = p; p += (size_t)R * N * D;
    float* ssrc = p; p += (size_t)R * N * HH;
    float* sdst = p; p += (size_t)N * HH;
    float* den  = p; p += (size_t)N * HH;
    unsigned* menc = (unsigned*)p;

    hipMemsetAsync(agg,  0, (size_t)N * D * sizeof(float), stream);
    hipMemsetAsync(den,  0, (size_t)N * HH * sizeof(float), stream);
    hipMemsetAsync(menc, 0, (size_t)N * HH * sizeof(unsigned), stream);

    pool_kernel<<<N, D, 0, stream>>>(feat, mask, spk, spk_emb, x, N, L, D);

    dim3 gg(304, R + 1);
    gemm_wmma_kernel<<<gg, 256, 0, stream>>>(x, W_r, W_q, hr, q, N, D, R);

    int nh = N * HH;
    node_score_kernel<<<(nh + 255) / 256, 256, 0, stream>>>(hr, q, a_src, a_dst, ssrc, sdst, N, D, R);

    int eh = E * HH;
    edge_max_kernel<<<(eh + 255) / 256, 256, 0, stream>>>(ei, et, ssrc, sdst, menc, E, N);
    edge_den_kernel<<<(eh + 255) / 256, 256, 0, stream>>>(ei, et, ssrc, sdst, menc, den, E, N);

    int ew = E * 32;
    edge_agg_kernel<<<(ew + 255) / 256, 256, 0, stream>>>(ei, et, ssrc, sdst, menc, den, hr, agg, E, N, D);

    gemm_out_kernel<<<304, 256, 0, stream>>>(agg, W_out, x, b_out, out, N, D);
}